// Loss_82935818486116
// MI455X (gfx1250) — compile-verified
//
#include <hip/hip_runtime.h>
#include <math.h>

// Problem constants (from reference)
#define DIM      64
#define NQ       6
#define BATCH    4096
#define FLATN    2080          // 64*65/2
#define KSQ      6             // scaling-and-squaring doublings: theta = M / 2^6
#define EPSV     1e-12f
#define LDA      66            // padded LDS row stride (floats) -> conflict-free column reads
#define MATF     (DIM * LDA)   // floats per padded 64x64 matrix
#define NTHREADS 512           // 16 wave32s: one 16x16 C-tile per wave per matmul

typedef float v2f __attribute__((ext_vector_type(2)));
typedef float v8f __attribute__((ext_vector_type(8)));

// C[64x64] = alpha * A @ B  (+ C if accumulate), all padded-stride LDS matrices.
// 16 waves each own one 16x16 tile of C. Uses V_WMMA_F32_16X16X4_F32.
// ISA layouts (CDNA5 ISA):
//   A 16x4:  lanes 0-15 -> M=lane, VGPR0=K0,VGPR1=K1 ; lanes 16-31 -> K2,K3
//   B 4x16:  VGPR r -> row K=r (lanes 0-15), K=r+2 (lanes 16-31), N=lane&15
//   C/D:     VGPR r -> (M=r, N=lane) lanes 0-15 ; (M=r+8) lanes 16-31
__device__ __forceinline__ void mm64(float* __restrict__ C,
                                     const float* __restrict__ A,
                                     const float* __restrict__ B,
                                     float alpha, int accumulate) {
  const int tid  = threadIdx.x;
  const int wave = tid >> 5;
  const int lane = tid & 31;
  const int tm = (wave >> 2) << 4;   // tile row  (0,16,32,48)
  const int tn = (wave &  3) << 4;   // tile col
  const int l  = lane & 15;
  const int h  = lane >> 4;          // 0: K0/K1 half, 1: K2/K3 half

  v8f acc = {0.f, 0.f, 0.f, 0.f, 0.f, 0.f, 0.f, 0.f};
  const float* arow = A + (tm + l) * LDA;   // this lane's A row
  const float* bcol = B + tn + l;           // this lane's B column

#pragma unroll
  for (int k0 = 0; k0 < DIM; k0 += 4) {
    const int k = k0 + (h << 1);
    v2f a, b;
    a.x = arow[k];
    a.y = arow[k + 1];
    b.x = bcol[k * LDA];
    b.y = bcol[(k + 1) * LDA];
    acc = __builtin_amdgcn_wmma_f32_16x16x4_f32(
        /*neg_a=*/false, a, /*neg_b=*/false, b,
        /*c_mod=*/(short)0, acc, /*reuse_a=*/false, /*reuse_b=*/false);
  }

  float* crow = C + (tm + (h << 3)) * LDA + tn + l;
#pragma unroll
  for (int r = 0; r < 8; ++r) {
    float v = alpha * acc[r];
    if (accumulate) v += crow[r * LDA];
    crow[r * LDA] = v;
  }
}

// Single-qubit gate tables: 0 = H, 1 = Y-basis (H @ S_dag), 2 = I
__constant__ float g_mre[3][4] = {
    {0.70710678f,  0.70710678f, 0.70710678f, -0.70710678f},  // H
    {0.70710678f,  0.f,         0.70710678f,  0.f        },  // YH: [[s,-is],[s,is]]
    {1.f,          0.f,         0.f,          1.f        }}; // I
__constant__ float g_mim[3][4] = {
    {0.f, 0.f,          0.f, 0.f        },
    {0.f, -0.70710678f, 0.f, 0.70710678f},
    {0.f, 0.f,          0.f, 0.f        }};

__global__ __launch_bounds__(NTHREADS)
void qloss_kernel(const float* __restrict__ pflat,
                  const float* __restrict__ time,
                  const float* __restrict__ rho_in,
                  const int*   __restrict__ tgt,
                  const int*   __restrict__ basis,
                  float*       __restrict__ out) {
  __shared__ float sm[6 * MATF + DIM];
  float* b0 = sm;             // theta -> R_re -> rho
  float* b1 = sm + 1 * MATF;  // T2 -> S (sin)
  float* b2 = sm + 2 * MATF;  // T4 -> tmp -> R_im -> G_re
  float* b3 = sm + 3 * MATF;  // T6 -> tmp -> W_re
  float* b4 = sm + 4 * MATF;  // C (cos) -> G_im
  float* b5 = sm + 5 * MATF;  // sin inner poly -> W_im
  float* probs = sm + 6 * MATF;

  const int bidx = blockIdx.x;
  const int tid  = threadIdx.x;
  const float t  = time[bidx];
  const float scale = 0.25f * t * (1.f / 64.f);   // theta = DOWNSCALING*t*H / 2^KSQ

  // ---- build theta (symmetric) from packed lower triangle ----
  const float* flat = pflat + (size_t)bidx * FLATN;
  for (int f = tid; f < FLATN; f += NTHREADS) {
    int i = (int)((sqrtf(8.f * (float)f + 1.f) - 1.f) * 0.5f);
    while ((i + 1) * (i + 2) / 2 <= f) ++i;
    while (i * (i + 1) / 2 > f) --i;
    const int j = f - i * (i + 1) / 2;
    const float v = scale * flat[f];
    b0[i * LDA + j] = v;
    b0[j * LDA + i] = v;
  }
  __syncthreads();

  // ---- Taylor base: powers of theta ----
  mm64(b1, b0, b0, 1.f, 0); __syncthreads();   // T2 = theta^2
  mm64(b2, b1, b1, 1.f, 0); __syncthreads();   // T4
  mm64(b3, b2, b1, 1.f, 0); __syncthreads();   // T6
  for (int e = tid; e < DIM * DIM; e += NTHREADS) {
    const int i = e >> 6, j = e & 63, o = i * LDA + j;
    const float t2 = b1[o], t4 = b2[o], t6 = b3[o];
    const float id = (i == j) ? 1.f : 0.f;
    b4[o] = id - 0.5f * t2 + (1.f / 24.f) * t4 - (1.f / 720.f) * t6;        // cos(theta)
    b5[o] = id - (1.f / 6.f) * t2 + (1.f / 120.f) * t4 - (1.f / 5040.f) * t6; // sin poly
  }
  __syncthreads();
  mm64(b1, b0, b5, 1.f, 0); __syncthreads();   // S = theta @ poly = sin(theta)

  // ---- 6 angle doublings: cos2x = 2c^2 - I, sin2x = 2 s c ----
  for (int it = 0; it < KSQ; ++it) {
    mm64(b2, b4, b4, 1.f, 0);                  // c^2
    mm64(b3, b1, b4, 1.f, 0);                  // s*c
    __syncthreads();
    for (int e = tid; e < DIM * DIM; e += NTHREADS) {
      const int i = e >> 6, j = e & 63, o = i * LDA + j;
      b4[o] = 2.f * b2[o] - ((i == j) ? 1.f : 0.f);
      b1[o] = 2.f * b3[o];
    }
    __syncthreads();
  }
  // Now U = exp(-i M) = b4 - i*b1   (U_re = cos M, U_im = -sin M)

  // ---- build R = kron of 6 single-qubit gates ----
  int bq[NQ];
#pragma unroll
  for (int q = 0; q < NQ; ++q) bq[q] = basis[bidx * NQ + q];
  for (int e = tid; e < DIM * DIM; e += NTHREADS) {
    const int i = e >> 6, j = e & 63, o = i * LDA + j;
    float re = 1.f, im = 0.f;
#pragma unroll
    for (int q = 0; q < NQ; ++q) {
      const int iq = (i >> (5 - q)) & 1;
      const int jq = (j >> (5 - q)) & 1;
      const int m  = bq[q];
      const float cr = g_mre[m][iq * 2 + jq];
      const float ci = g_mim[m][iq * 2 + jq];
      const float nr = re * cr - im * ci;
      const float ni = re * ci + im * cr;
      re = nr; im = ni;
    }
    b0[o] = re;   // R_re
    b2[o] = im;   // R_im
  }
  __syncthreads();

  // ---- W = R @ U : W_re = Rre@C + Rim@S ; W_im = Rim@C - Rre@S ----
  mm64(b3, b0, b4, 1.f, 0);                    // W_re  = Rre*C
  mm64(b5, b2, b4, 1.f, 0);                    // W_im  = Rim*C
  mm64(b3, b2, b1, 1.f, 1);                    // W_re += Rim*S
  mm64(b5, b0, b1, -1.f, 1);                   // W_im -= Rre*S
  __syncthreads();

  // ---- load rho, G = W @ rho ----
  const float* rp = rho_in + (size_t)bidx * DIM * DIM;
  for (int e = tid; e < DIM * DIM; e += NTHREADS)
    b0[(e >> 6) * LDA + (e & 63)] = rp[e];
  __syncthreads();
  mm64(b2, b3, b0, 1.f, 0);                    // G_re = W_re @ rho
  mm64(b4, b5, b0, 1.f, 0);                    // G_im = W_im @ rho
  __syncthreads();

  // ---- diag_i = sum_k G[i,k] * conj(W[i,k]) ; p_i = |diag_i|^2 ----
  const int row = tid >> 3;
  const int c0  = tid & 7;
  float dre = 0.f, dim = 0.f;
  for (int k = c0; k < DIM; k += 8) {
    const int o = row * LDA + k;
    const float gre = b2[o], gim = b4[o];
    const float wre = b3[o], wim = b5[o];
    dre += gre * wre + gim * wim;
    dim += gim * wre - gre * wim;
  }
#pragma unroll
  for (int off = 4; off >= 1; off >>= 1) {
    dre += __shfl_xor(dre, off, 32);
    dim += __shfl_xor(dim, off, 32);
  }
  if (c0 == 0) probs[row] = dre * dre + dim * dim;
  __syncthreads();

  if (tid == 0) {
    float s = 0.f;
#pragma unroll 8
    for (int k = 0; k < DIM; ++k) s += probs[k];
    float tp = probs[tgt[bidx]] / s;
    tp = fmaxf(tp, EPSV);
    atomicAdd(out, -logf(tp) * (1.f / (6.f * (float)BATCH)));
  }
}

extern "C" void kernel_launch(void* const* d_in, const int* in_sizes, int n_in,
                              void* d_out, int out_size, void* d_ws, size_t ws_size,
                              hipStream_t stream) {
  const float* pflat  = (const float*)d_in[0];  // (4096, 2080) f32
  const float* timev  = (const float*)d_in[1];  // (4096,)      f32
  const float* rho    = (const float*)d_in[2];  // (4096,64,64) f32
  const int*   tgt    = (const int*)d_in[3];    // (4096,)      i32
  const int*   basis  = (const int*)d_in[4];    // (4096,6)     i32
  float* out = (float*)d_out;

  hipMemsetAsync(out, 0, sizeof(float), stream);
  hipLaunchKernelGGL(qloss_kernel, dim3(BATCH), dim3(NTHREADS), 0, stream,
                     pflat, timev, rho, tgt, basis, out);
}